// RNF_GATConv_18253611008263
// MI455X (gfx1250) — compile-verified
//
#include <hip/hip_runtime.h>
#include <hip/hip_bf16.h>
#include <cstddef>

#define NEG_SLOPE 0.2f
#define TANH_SCALE 10000.0f

typedef __attribute__((ext_vector_type(2))) float v2f;
typedef __attribute__((ext_vector_type(8))) float v8f;

// ---------------------------------------------------------------------------
// GEMM: Y[N, ncols] = A[N, K] * W[ncols, K]^T  using V_WMMA_F32_16X16X4_F32.
// 256 threads (8 waves). Wave (wr, wc) owns output tile rows [wr*16,+16) x
// cols [wc*16,+16). A-tile staged in LDS with +4 float row padding (stride
// 260 floats -> lanes hit distinct banks). W is small (<=128KB) and stays
// L2/L0 resident, read directly as float2 fragments.
// ---------------------------------------------------------------------------
__global__ void gemm_wmma_f32(const float* __restrict__ A,
                              const float* __restrict__ W,
                              float* __restrict__ Y,
                              int N, int K, int ncols,
                              int ncoltiles, int rowtiles) {
    extern __shared__ float ldsA[];
    const int KP    = K + 4;
    const int tid   = threadIdx.x;
    const int wave  = tid >> 5;
    const int lane  = tid & 31;
    const int l16   = lane & 15;
    const int half  = lane >> 4;               // 0: K=0,1  1: K=2,3
    const int rowsPerBlk = rowtiles * 16;
    const int tiles = ncoltiles * rowtiles;
    const int wc = wave % ncoltiles;
    const int wr = wave / ncoltiles;
    const int numRowBlks = (N + rowsPerBlk - 1) / rowsPerBlk;

    for (int rb = blockIdx.x; rb < numRowBlks; rb += gridDim.x) {
        const int row0 = rb * rowsPerBlk;
        __syncthreads();   // protect LDS from previous iteration's readers
        // Stage A rows into LDS (float4, padded row stride KP)
        const int aElems = rowsPerBlk * K;
        for (int i = tid * 4; i < aElems; i += blockDim.x * 4) {
            const int r = i / K;
            const int c = i - r * K;
            float4 v = make_float4(0.f, 0.f, 0.f, 0.f);
            if (row0 + r < N)
                v = *(const float4*)(A + (size_t)(row0 + r) * K + c);
            *(float4*)(ldsA + r * KP + c) = v;
        }
        __syncthreads();

        if (wave < tiles) {
            v8f acc = {0.f, 0.f, 0.f, 0.f, 0.f, 0.f, 0.f, 0.f};
            // A frag: lane l16 -> M row (wr*16+l16); half selects K pair.
            const float* aBase = ldsA + (wr * 16 + l16) * KP + 2 * half;
            // B frag: B[k][n] = W[wc*16 + n][k]; lane l16 -> N col; half -> K pair.
            const float* bBase = W + (size_t)(wc * 16 + l16) * K + 2 * half;
            for (int kk = 0; kk < K; kk += 4) {
                v2f av = *(const v2f*)(aBase + kk);
                v2f bv = *(const v2f*)(bBase + kk);
                acc = __builtin_amdgcn_wmma_f32_16x16x4_f32(
                    /*neg_a=*/false, av, /*neg_b=*/false, bv,
                    /*c_mod=*/(short)0, acc,
                    /*reuse_a=*/false, /*reuse_b=*/false);
            }
            // C/D layout: VGPR r -> M = r + 8*half, N = l16
            const int gm0 = row0 + wr * 16 + 8 * half;
            const int col = wc * 16 + l16;
#pragma unroll
            for (int r = 0; r < 8; ++r) {
                const int gr = gm0 + r;
                if (gr < N) Y[(size_t)gr * ncols + col] = acc[r];
            }
        }
    }
}

// ---------------------------------------------------------------------------
// Per-node scalars: Ls/Rd (attention branch, leaky+proj folded per node),
// Ts/Td (tanh branch projections).
// ---------------------------------------------------------------------------
__global__ void node_scalars(const float* __restrict__ Y1,   // [N, H*D]
                             const float* __restrict__ Y2,   // [N, D]
                             const float* __restrict__ attn_l,
                             const float* __restrict__ attn_r,
                             const float* __restrict__ wpa,   // [2D]
                             const float* __restrict__ wpt,   // [2D]
                             float* __restrict__ Ls, float* __restrict__ Rd,
                             float* __restrict__ Ts, float* __restrict__ Td,
                             int N, int H, int D) {
    int i = blockIdx.x * blockDim.x + threadIdx.x;
    if (i >= N * H) return;
    int n = i / H, h = i - n * H;
    const float* y  = Y1 + ((size_t)n * H + h) * D;
    const float* al = attn_l + h * D;
    const float* ar = attn_r + h * D;
    float ls = 0.f, rd = 0.f;
    for (int k = 0; k < D; ++k) {
        float v  = y[k];
        float el = v * al[k];
        float er = v * ar[k];
        ls += (el >= 0.f ? el : NEG_SLOPE * el) * wpa[k];
        rd += (er >= 0.f ? er : NEG_SLOPE * er) * wpa[D + k];
    }
    Ls[i] = ls;
    Rd[i] = rd;
    if (h == 0) {
        const float* y2 = Y2 + (size_t)n * D;
        float ts = 0.f, td = 0.f;
        for (int k = 0; k < D; ++k) {
            float v = y2[k];
            ts += v * wpt[k];
            td += v * wpt[D + k];
        }
        Ts[n] = ts;
        Td[n] = td;
    }
}

// Init: rst = bias broadcast, m = -inf, s = 0
__global__ void init_buffers(float* __restrict__ rst, const float* __restrict__ bias,
                             float* __restrict__ m, float* __restrict__ s,
                             int N, int H, int HD) {
    int i = blockIdx.x * blockDim.x + threadIdx.x;
    if (i < N * HD) rst[i] = bias[i % HD];
    if (i < N * H) { m[i] = -__builtin_inff(); s[i] = 0.f; }
}

__device__ inline void atomicMaxF(float* addr, float v) {
    if (__float_as_int(v) >= 0)
        atomicMax((int*)addr, __float_as_int(v));
    else
        atomicMin((unsigned int*)addr, __float_as_uint(v));
}

// logits[e,h] = Ls[src,h] + Rd[dst,h];  segment max over dst
__global__ void edge_logits_max(const int* __restrict__ src, const int* __restrict__ dst,
                                const float* __restrict__ Ls, const float* __restrict__ Rd,
                                float* __restrict__ logits, float* __restrict__ m,
                                int E, int H) {
    int e = blockIdx.x * blockDim.x + threadIdx.x;
    if (e >= E) return;
    int sn = src[e], dn = dst[e];
    for (int h = 0; h < H; ++h) {
        float lg = Ls[sn * H + h] + Rd[dn * H + h];
        logits[(size_t)e * H + h] = lg;
        atomicMaxF(&m[dn * H + h], lg);
    }
}

// e = exp(logit - m[dst]); segment sum over dst; overwrite logits with e
__global__ void edge_exp_sum(const int* __restrict__ dst, const float* __restrict__ m,
                             float* __restrict__ logits, float* __restrict__ s,
                             int E, int H) {
    int e = blockIdx.x * blockDim.x + threadIdx.x;
    if (e >= E) return;
    int dn = dst[e];
    for (int h = 0; h < H; ++h) {
        float ev = expf(logits[(size_t)e * H + h] - m[dn * H + h]);
        logits[(size_t)e * H + h] = ev;
        atomicAdd(&s[dn * H + h], ev);
    }
}

// One wave per edge: coef = (sum_h alpha) * tanh(scale*(Ts[src]+Td[dst]));
// rst[dst,:] += target[src,:] * coef  (lane-strided, coalesced 128B/step)
__global__ void edge_aggregate(const int* __restrict__ src, const int* __restrict__ dst,
                               const float* __restrict__ expv, const float* __restrict__ s,
                               const float* __restrict__ Ts, const float* __restrict__ Td,
                               const float* __restrict__ target, float* __restrict__ rst,
                               int E, int H, int HD) {
    int gw   = (blockIdx.x * blockDim.x + threadIdx.x) >> 5;
    int lane = threadIdx.x & 31;
    if (gw >= E) return;
    int sn = src[gw], dn = dst[gw];
    float a = 0.f;
    for (int h = 0; h < H; ++h)
        a += expv[(size_t)gw * H + h] / s[dn * H + h];
    float gate = tanhf(TANH_SCALE * (Ts[sn] + Td[dn]));
    float coef = a * gate;
    const float* trow = target + (size_t)sn * HD;
    float* orow = rst + (size_t)dn * HD;
    for (int j = lane; j < HD; j += 32)
        atomicAdd(&orow[j], trow[j] * coef);
}

// ---------------------------------------------------------------------------
extern "C" void kernel_launch(void* const* d_in, const int* in_sizes, int n_in,
                              void* d_out, int out_size, void* d_ws, size_t ws_size,
                              hipStream_t stream) {
    const float* feat    = (const float*)d_in[0];
    const float* target  = (const float*)d_in[1];
    const int*   src     = (const int*)d_in[2];
    const int*   dst     = (const int*)d_in[3];
    const float* W_attn  = (const float*)d_in[4];
    const float* W_tanh  = (const float*)d_in[5];
    const float* wpa     = (const float*)d_in[6];
    const float* wpt     = (const float*)d_in[7];
    const float* attn_l  = (const float*)d_in[8];
    const float* attn_r  = (const float*)d_in[9];
    const float* bias    = (const float*)d_in[10];
    float* rst = (float*)d_out;

    const int D  = in_sizes[6] / 2;            // 32
    const int F  = in_sizes[5] / D;            // 256
    const int HD = in_sizes[4] / F;            // 128
    const int H  = HD / D;                     // 4
    const int N  = in_sizes[0] / F;            // 50000
    const int E  = in_sizes[2];                // 800000

    // Workspace carve-out (256B aligned)
    size_t off = 0;
    auto carve = [&](size_t elems) {
        float* p = (float*)((char*)d_ws + off);
        off += ((elems * sizeof(float) + 255) / 256) * 256;
        return p;
    };
    float* Y1     = carve((size_t)N * HD);   // fc_attn output
    float* Y2     = carve((size_t)N * D);    // fc_tanh output
    float* Ls     = carve((size_t)N * H);
    float* Rd     = carve((size_t)N * H);
    float* Ts     = carve((size_t)N);
    float* Td     = carve((size_t)N);
    float* m      = carve((size_t)N * H);
    float* ssum   = carve((size_t)N * H);
    float* logits = carve((size_t)E * H);    // reused as exp values
    (void)ws_size; (void)out_size; (void)n_in;

    const int KP = F + 4;

    // GEMM 1: Y1 = feat @ W_attn.T   (ncols=128 -> 8 col tiles x 1 row tile)
    {
        int ncoltiles = HD / 16, rowtiles = 1;
        int rowsPerBlk = rowtiles * 16;
        int nrb = (N + rowsPerBlk - 1) / rowsPerBlk;
        int grid = nrb < 1024 ? nrb : 1024;
        size_t lds = (size_t)rowsPerBlk * KP * sizeof(float);   // ~16.6 KB
        gemm_wmma_f32<<<grid, 256, lds, stream>>>(feat, W_attn, Y1, N, F, HD,
                                                  ncoltiles, rowtiles);
    }
    // GEMM 2: Y2 = feat @ W_tanh.T   (ncols=32 -> 2 col tiles x 3 row tiles)
    {
        int ncoltiles = D / 16, rowtiles = 3;
        int rowsPerBlk = rowtiles * 16;
        int nrb = (N + rowsPerBlk - 1) / rowsPerBlk;
        int grid = nrb < 1024 ? nrb : 1024;
        size_t lds = (size_t)rowsPerBlk * KP * sizeof(float);   // ~49.9 KB
        gemm_wmma_f32<<<grid, 256, lds, stream>>>(feat, W_tanh, Y2, N, F, D,
                                                  ncoltiles, rowtiles);
    }
    // Per-node scalars
    {
        int total = N * H;
        node_scalars<<<(total + 255) / 256, 256, 0, stream>>>(
            Y1, Y2, attn_l, attn_r, wpa, wpt, Ls, Rd, Ts, Td, N, H, D);
    }
    // Init output (bias broadcast) + softmax accumulators
    {
        int total = N * HD;
        init_buffers<<<(total + 255) / 256, 256, 0, stream>>>(rst, bias, m, ssum, N, H, HD);
    }
    // Edge phase
    edge_logits_max<<<(E + 255) / 256, 256, 0, stream>>>(src, dst, Ls, Rd, logits, m, E, H);
    edge_exp_sum<<<(E + 255) / 256, 256, 0, stream>>>(dst, m, logits, ssum, E, H);
    {
        long long threads = (long long)E * 32;
        int grid = (int)((threads + 255) / 256);
        edge_aggregate<<<grid, 256, 0, stream>>>(src, dst, logits, ssum, Ts, Td,
                                                 target, rst, E, H, HD);
    }
}